// AxonalConnections_57105885167626
// MI455X (gfx1250) — compile-verified
//
#include <hip/hip_runtime.h>

typedef __attribute__((ext_vector_type(2))) float v2f;
typedef __attribute__((ext_vector_type(8))) float v8f;

#define GRID_DIM 256
#define PSIZE    16
#define S_DIM    256
#define T_DIM    256
#define TEMP_INV 10.0f
#define LN_EPS   1e-5f

__device__ __forceinline__ float block_sum(float v, float* red, int tid) {
    red[tid] = v; __syncthreads();
    #pragma unroll
    for (int off = 128; off > 0; off >>= 1) {
        if (tid < off) red[tid] += red[tid + off];
        __syncthreads();
    }
    float r = red[0]; __syncthreads();
    return r;
}

__device__ __forceinline__ float block_max(float v, float* red, int tid) {
    red[tid] = v; __syncthreads();
    #pragma unroll
    for (int off = 128; off > 0; off >>= 1) {
        if (tid < off) red[tid] = fmaxf(red[tid], red[tid + off]);
        __syncthreads();
    }
    float r = red[0]; __syncthreads();
    return r;
}

__global__ __launch_bounds__(256) void axonal_connections_kernel(
    const float* __restrict__ spikes,   // [B,256,256]
    const float* __restrict__ Wdyn,     // [B,N,T,S]
    const float* __restrict__ gamma,    // [T]
    const float* __restrict__ beta,     // [T]
    const float* __restrict__ gates,    // [N]
    const float* __restrict__ biases,   // [N]
    float* __restrict__ out)            // [B,256,256]
{
    __shared__ float sp_lds[S_DIM];
    __shared__ float out_lds[T_DIM];
    __shared__ float red[256];

    const int tid = threadIdx.x;
    const int bn  = blockIdx.x;          // b*256 + n
    const int b   = bn >> 8;
    const int n   = bn & 255;
    const int py  = n >> 4;
    const int px  = n & 15;

    // Unfold: sp[s = iy*16+ix] = spikes[b, py*16+iy, px*16+ix]
    {
        int iy = tid >> 4, ix = tid & 15;
        sp_lds[tid] = spikes[(size_t)b * GRID_DIM * GRID_DIM +
                             (size_t)(py * PSIZE + iy) * GRID_DIM +
                             (px * PSIZE + ix)];
    }
    __syncthreads();

    const int wave = tid >> 5;
    const int lane = tid & 31;
    const int h    = lane >> 4;   // half-wave (selects K pair)
    const int j    = lane & 15;   // output column within 16

    const size_t Wbase = (size_t)bn * T_DIM * S_DIM;

    // Each wave produces 2 groups of 16 outputs: out[g*16 + j]
    #pragma unroll
    for (int gi = 0; gi < 2; ++gi) {
        const int g    = wave * 2 + gi;
        const int trow = g * 16 + j;
        // B operand: W row (g*16+j), 2 consecutive f32 per lane per k-step
        const v2f* wp = (const v2f*)(Wdyn + Wbase + (size_t)trow * S_DIM) + h;
        // A operand: spike vector replicated across the 16 M-rows
        const v2f* ap = (const v2f*)sp_lds + h;
        v8f c = {};
        #pragma unroll 4
        for (int ks = 0; ks < 64; ++ks) {
            v2f bvec = __builtin_nontemporal_load(wp + ks * 2);  // stream-once: NT hint
            v2f avec = ap[ks * 2];
            c = __builtin_amdgcn_wmma_f32_16x16x4_f32(
                    /*neg_a=*/false, avec, /*neg_b=*/false, bvec,
                    /*c_mod=*/(short)0, c, /*reuse_a=*/false, /*reuse_b=*/false);
        }
        // All D rows identical (A rows replicated); lanes 0..15 hold M=0, N=lane
        if (lane < 16) out_lds[g * 16 + lane] = c[0];
    }
    __syncthreads();

    // LayerNorm over T (biased variance, matching jnp.var)
    float val  = out_lds[tid];
    float mean = block_sum(val, red, tid) * (1.0f / 256.0f);
    float d    = val - mean;
    float var  = block_sum(d * d, red, tid) * (1.0f / 256.0f);
    float logit = (d * rsqrtf(var + LN_EPS) * gamma[tid] + beta[tid]) * TEMP_INV;

    // Temperature softmax over T (max-stabilized)
    float m  = block_max(logit, red, tid);
    float e  = __expf(logit - m);
    float se = block_sum(e, red, tid);
    float sm = e / se;

    // Scalar affine per patch: gate * sum(sp) + bias
    float ssum = block_sum(sp_lds[tid], red, tid);
    float scal = gates[n] * ssum + biases[n];

    // Fold: pixel (py*16+ty, px*16+tx) <- final[b,n,t]
    int ty = tid >> 4, tx = tid & 15;
    out[(size_t)b * GRID_DIM * GRID_DIM +
        (size_t)(py * PSIZE + ty) * GRID_DIM +
        (px * PSIZE + tx)] = sm * scal;
}

extern "C" void kernel_launch(void* const* d_in, const int* in_sizes, int n_in,
                              void* d_out, int out_size, void* d_ws, size_t ws_size,
                              hipStream_t stream) {
    const float* spikes = (const float*)d_in[0];
    const float* Wdyn   = (const float*)d_in[1];
    const float* gamma  = (const float*)d_in[2];
    const float* beta   = (const float*)d_in[3];
    const float* gates  = (const float*)d_in[4];
    const float* biases = (const float*)d_in[5];
    float* out = (float*)d_out;

    dim3 grid(4 * 256);   // one block per (b, n)
    dim3 block(256);      // 8 wave32s
    hipLaunchKernelGGL(axonal_connections_kernel, grid, block, 0, stream,
                       spikes, Wdyn, gamma, beta, gates, biases, out);
}